// DynamicMessagePassing_69810398429288
// MI455X (gfx1250) — compile-verified
//
#include <hip/hip_runtime.h>

typedef __attribute__((ext_vector_type(2))) float v2f;
typedef __attribute__((ext_vector_type(8))) float v8f;

constexpr int kN   = 8;
constexpr int kC   = 256;
constexpr int kHW  = 128 * 128;          // 16384
constexpr int kCHW = kC * kHW;
constexpr int kS   = 16;

// ---------------------------------------------------------------------------
// Stage 1: gather S_rgb[n][s][c] = rgb[n, c, idx[n][s]] and h = S_rgb @ W^T + b
// One block per batch. Tiny (16.8 MFLOP total).
// ---------------------------------------------------------------------------
__global__ __launch_bounds__(256)
void dmp_stage1(const float* __restrict__ rgb,
                const long long* __restrict__ idx,   // reference dtype: int64
                const float* __restrict__ W,
                const float* __restrict__ bias,
                float* __restrict__ S_ws,            // [N][S][C]
                float* __restrict__ h_ws) {          // [N][S][C]
  __shared__ float Sl[kS][kC];
  const int n = blockIdx.x;
  const int t = threadIdx.x;                         // t = channel
  for (int s = 0; s < kS; ++s) {
    const int q = (int)idx[n * kS + s];
    const float v = rgb[n * kCHW + t * kHW + q];
    Sl[s][t] = v;
    S_ws[(n * kS + s) * kC + t] = v;
  }
  __syncthreads();
  float acc[kS];
  const float bb = bias[t];
#pragma unroll
  for (int s = 0; s < kS; ++s) acc[s] = bb;
  for (int cc = 0; cc < kC; ++cc) {
    const float w = W[t * kC + cc];                  // h[s][d] = sum_c S[s][c]*W[d][c]
#pragma unroll
    for (int s = 0; s < kS; ++s) acc[s] += Sl[s][cc] * w;
  }
  for (int s = 0; s < kS; ++s) h_ws[(n * kS + s) * kC + t] = acc[s];
}

// ---------------------------------------------------------------------------
// Stage 2: fused adj = x@S^T ; out = adj@h ; relu(identity + gamma*out)
// 256 threads = 8 waves; each wave owns one 16-row q-tile. Full fp32 WMMA.
// ---------------------------------------------------------------------------
__global__ __launch_bounds__(256)
void dmp_stage2(const float* __restrict__ rgb,
                const float* __restrict__ S_ws,
                const float* __restrict__ h_ws,
                const float* __restrict__ gamma,
                float* __restrict__ out) {
  __shared__ float Sl[kS * 258];        // S_rgb [s][c], pitch 258 (bank-spread, 8B aligned pairs)
  __shared__ float hT[kC * 18];         // h^T   [c][s], pitch 18  (bank-spread, 8B aligned pairs)
  __shared__ float adjl[8][kS * 17];    // per-wave adj^T [s][q], pitch 17

  const int n = blockIdx.y;
  const int t = threadIdx.x;

  // cooperative LDS fill (coalesced global reads)
  for (int i = 0; i < kS; ++i) {
    const int id = i * 256 + t;
    const int s = id >> 8, cc = id & 255;
    Sl[s * 258 + cc] = S_ws[n * kS * kC + id];
    hT[cc * 18 + s]  = h_ws[n * kS * kC + id];
  }
  __syncthreads();

  const float g = gamma[0];
  const int wave = t >> 5, lane = t & 31;
  const int ll = lane & 15, hh = lane >> 4;
  const int qb = (blockIdx.x * 8 + wave) * kS;

  // x[q][k] = rgb[n, k, qb+q]  -> lanes 0..15 cover consecutive q: coalesced
  const float* xb = rgb + (size_t)n * kCHW + qb + ll;

  // ---- GEMM A: adjT[16s x 16q] = S_rgb[16 x 256k] @ x^T[256k x 16q] -------
  v8f acc = {};
#pragma unroll 4
  for (int k0 = 0; k0 < kC; k0 += 4) {
    const int kk = k0 + 2 * hh;                        // A/B K-split per lane half
    v2f a = *(const v2f*)&Sl[ll * 258 + kk];           // A[M=s=ll][K=kk,kk+1]
    v2f bf;
    bf.x = xb[kk * kHW];                               // B[K=kk ][N=q=ll]
    bf.y = xb[(kk + 1) * kHW];                         // B[K=kk+1][N=q=ll]
    acc = __builtin_amdgcn_wmma_f32_16x16x4_f32(
        false, a, false, bf, (short)0, acc, false, false);
  }

  // reshape adjT from C/D layout into LDS so it can be re-read as a B operand
  float* aw = &adjl[wave][0];
#pragma unroll
  for (int j = 0; j < 8; ++j)
    aw[(j + 8 * hh) * 17 + ll] = acc[j];               // adjT[s = j+8*hh][q = ll]

  // ---- GEMM B: out[16c x 16q] = h^T[16c x 16s] @ adjT[16s x 16q] ----------
  for (int ct = 0; ct < kC; ct += 16) {
    v8f acc2 = {};
#pragma unroll
    for (int ks = 0; ks < kS; ks += 4) {
      const int kk = ks + 2 * hh;
      v2f a2 = *(const v2f*)&hT[(ct + ll) * 18 + kk];  // A[M=c][K=s]
      v2f b2;
      b2.x = aw[kk * 17 + ll];                         // B[K=s][N=q]
      b2.y = aw[(kk + 1) * 17 + ll];
      acc2 = __builtin_amdgcn_wmma_f32_16x16x4_f32(
          false, a2, false, b2, (short)0, acc2, false, false);
    }
    // epilogue: identity + gamma*out, relu; C/D layout == NCHW coalesced
#pragma unroll
    for (int j = 0; j < 8; ++j) {
      const int cc = ct + j + 8 * hh;
      const size_t o = (size_t)n * kCHW + (size_t)cc * kHW + qb + ll;
      const float v = rgb[o] + g * acc2[j];
      out[o] = v > 0.f ? v : 0.f;
    }
  }
}

// ---------------------------------------------------------------------------
extern "C" void kernel_launch(void* const* d_in, const int* in_sizes, int n_in,
                              void* d_out, int out_size, void* d_ws, size_t ws_size,
                              hipStream_t stream) {
  const float*     rgb   = (const float*)d_in[0];
  const long long* idx   = (const long long*)d_in[1];  // jnp.int64 indices
  const float*     W     = (const float*)d_in[2];
  const float*     bias  = (const float*)d_in[3];
  const float*     gamma = (const float*)d_in[4];
  float* out = (float*)d_out;

  float* S_ws = (float*)d_ws;                  // 8*16*256 floats
  float* h_ws = S_ws + kN * kS * kC;           // 8*16*256 floats

  dmp_stage1<<<kN, 256, 0, stream>>>(rgb, idx, W, bias, S_ws, h_ws);

  dim3 grid(kHW / (kS * 8), kN);               // 128 x 8 blocks, 8 waves each
  dmp_stage2<<<grid, 256, 0, stream>>>(rgb, S_ws, h_ws, gamma, out);
}